// BlockIDCT2D_59906203845016
// MI455X (gfx1250) — compile-verified
//
#include <hip/hip_runtime.h>

typedef float v2f __attribute__((ext_vector_type(2)));
typedef float v8f __attribute__((ext_vector_type(8)));

// Problem shape (from reference setup_inputs): x_dct (4,64,64,64,8,8) f32.
#define NBATCH 4
#define NCHAN  64
#define NBH    64
#define NBW    64
#define HOUT   512
#define WOUT   512

#define TILES_X (NBW / 2)              // 32 tiles of 2x2 blocks along width
#define TILES_Y (NBH / 2)              // 32 along height
#define WAVES_PER_BLOCK 8
#define LDS_STRIDE 20                  // floats: 16B-aligned rows, conflict-free columns

__global__ __launch_bounds__(256) void idct8x8_wmma_kernel(
    const float* __restrict__ x,       // (N,C,nbh,nbw,8,8)
    const float* __restrict__ idct,    // (8,8) row-major: M[n][k]
    float* __restrict__ out)           // (N,C,512,512)
{
    __shared__ float lds_in [WAVES_PER_BLOCK][16 * LDS_STRIDE];
    __shared__ float lds_mid[WAVES_PER_BLOCK][16 * LDS_STRIDE];

    const int tid  = threadIdx.x;
    const int wave = tid >> 5;
    const int lane = tid & 31;
    const int l16  = lane & 15;
    const int hi   = lane >> 4;        // 0: lanes 0-15, 1: lanes 16-31

    const int tile = blockIdx.x * WAVES_PER_BLOCK + wave;   // 0 .. 262143
    const int tj = tile % TILES_X;
    const int ti = (tile / TILES_X) % TILES_Y;
    const int ch = (tile / (TILES_X * TILES_Y)) % NCHAN;
    const int nb = tile / (TILES_X * TILES_Y * NCHAN);

    // ---- Build A operand: Md = diag(M, M) in 16x16x4 A layout ----
    // slice s, vgpr v, lane: A[M = l16, K = 4s + 2*hi + v]  (ISA 7.12.2, 32-bit 16x4 A)
    v2f a[4];
#pragma unroll
    for (int s = 0; s < 4; ++s) {
#pragma unroll
        for (int v = 0; v < 2; ++v) {
            const int m = l16;
            const int k = 4 * s + 2 * hi + v;
            a[s][v] = ((m >> 3) == (k >> 3)) ? idct[(m & 7) * 8 + (k & 7)] : 0.0f;
        }
    }

    // ---- Stage the 2x2 block group as a row-major 16x16 tile in LDS ----
    // Block (bi,bj) element (r,c) -> Xbig[bi*8+r, bj*8+c].
    // Lane covers one 8-float row: bi = hi, bj = l16/8, r8 = l16%8.
    const long base = ((((long)nb * NCHAN + ch) * NBH + 2 * ti) * NBW + 2 * tj) * 64;
    {
        const int bi = hi;
        const int bj = l16 >> 3;
        const int r8 = l16 & 7;
        const float* src = x + base + (long)bi * (NBW * 64) + bj * 64 + r8 * 8;
        float4 f0 = *(const float4*)(src);
        float4 f1 = *(const float4*)(src + 4);
        float* dst = &lds_in[wave][(bi * 8 + r8) * LDS_STRIDE + bj * 8];
        *(float4*)(dst)     = f0;
        *(float4*)(dst + 4) = f1;
    }
    // DS ops are in-order within a wave; this region is wave-private -> no barrier.

    // ---- Multiply 1: P = Md * Xbig^T  (B1[k,n] = Xbig[n,k], read transposed) ----
    v8f p = {0.f, 0.f, 0.f, 0.f, 0.f, 0.f, 0.f, 0.f};
#pragma unroll
    for (int s = 0; s < 4; ++s) {
        v2f b;
        b[0] = lds_in[wave][l16 * LDS_STRIDE + (4 * s + 2 * hi + 0)];
        b[1] = lds_in[wave][l16 * LDS_STRIDE + (4 * s + 2 * hi + 1)];
        p = __builtin_amdgcn_wmma_f32_16x16x4_f32(false, a[s], false, b,
                                                  (short)0, p, false, false);
    }

    // ---- Stage P (D layout: vgpr v -> row v + 8*hi, col l16) row-major into LDS ----
#pragma unroll
    for (int v = 0; v < 8; ++v)
        lds_mid[wave][(v + 8 * hi) * LDS_STRIDE + l16] = p[v];

    // ---- Multiply 2: Y = Md * P^T  (B2[k,n] = P[n,k]) ----
    v8f y = {0.f, 0.f, 0.f, 0.f, 0.f, 0.f, 0.f, 0.f};
#pragma unroll
    for (int s = 0; s < 4; ++s) {
        v2f b;
        b[0] = lds_mid[wave][l16 * LDS_STRIDE + (4 * s + 2 * hi + 0)];
        b[1] = lds_mid[wave][l16 * LDS_STRIDE + (4 * s + 2 * hi + 1)];
        y = __builtin_amdgcn_wmma_f32_16x16x4_f32(false, a[s], false, b,
                                                  (short)0, y, false, false);
    }

    // ---- Store: D layout maps directly onto the 16x16 raster tile ----
    // vgpr v, lane: out row 16*ti + v + 8*hi, col 16*tj + l16
    const long obase = (((long)nb * NCHAN + ch) * HOUT + 16 * ti) * (long)WOUT + 16 * tj;
#pragma unroll
    for (int v = 0; v < 8; ++v) {
        const int row = v + 8 * hi;
        out[obase + (long)row * WOUT + l16] = y[v];
    }
}

extern "C" void kernel_launch(void* const* d_in, const int* in_sizes, int n_in,
                              void* d_out, int out_size, void* d_ws, size_t ws_size,
                              hipStream_t stream) {
    const float* x    = (const float*)d_in[0];   // x_dct
    const float* idct = (const float*)d_in[1];   // idct_mat (8x8)
    float* out        = (float*)d_out;

    const int total_tiles = NBATCH * NCHAN * TILES_Y * TILES_X;   // 262144
    const int grid = total_tiles / WAVES_PER_BLOCK;               // 32768
    idct8x8_wmma_kernel<<<grid, 256, 0, stream>>>(x, idct, out);
}